// SimpleGCNNet_70824010711175
// MI455X (gfx1250) — compile-verified
//
#include <hip/hip_runtime.h>
#include <hip/hip_bf16.h>

#define C_DIM 256
#define EPSV  1e-12f

typedef __attribute__((ext_vector_type(2))) float v2f;
typedef __attribute__((ext_vector_type(8))) float v8f;

// ---------------- utility: zero a float buffer ----------------
__global__ __launch_bounds__(256)
void zero_kernel(float* __restrict__ p, int n) {
    int i = blockIdx.x * blockDim.x + threadIdx.x;
    if (i < n) p[i] = 0.0f;
}

// ---------------- L2-normalize rows: one wave32 per row ----------------
__global__ __launch_bounds__(256)
void normalize_rows(const float* __restrict__ x, float* __restrict__ y, int N) {
    int wave = (int)((blockIdx.x * blockDim.x + threadIdx.x) >> 5);
    int lane = threadIdx.x & 31;
    if (wave >= N) return;
    const float* xr = x + (size_t)wave * C_DIM;
    float v[8];
    float s = 0.0f;
#pragma unroll
    for (int i = 0; i < 8; ++i) {
        v[i] = xr[lane + 32 * i];
        s += v[i] * v[i];
    }
#pragma unroll
    for (int off = 16; off > 0; off >>= 1)
        s += __shfl_xor(s, off, 32);
    float nrm = sqrtf(s);
    float inv = 1.0f / fmaxf(nrm, EPSV);
    float* yr = y + (size_t)wave * C_DIM;
#pragma unroll
    for (int i = 0; i < 8; ++i)
        yr[lane + 32 * i] = v[i] * inv;
}

// ---------------- degree: deg[col[e]] += w[e] ----------------
__global__ __launch_bounds__(256)
void deg_kernel(const int* __restrict__ col, const float* __restrict__ w,
                float* __restrict__ deg, int E) {
    int e = blockIdx.x * blockDim.x + threadIdx.x;
    if (e < E) atomicAdd(&deg[col[e]], w[e]);
}

// ---------------- deg -> deg^{-1/2} (in place) ----------------
__global__ __launch_bounds__(256)
void dinv_kernel(float* __restrict__ deg, int N) {
    int i = blockIdx.x * blockDim.x + threadIdx.x;
    if (i < N) {
        float d = deg[i];
        deg[i] = (d > 0.0f) ? rsqrtf(d) : 0.0f;
    }
}

// ---------------- per-edge norm = dinv[row]*w*dinv[col] ----------------
__global__ __launch_bounds__(256)
void edge_norm_kernel(const int* __restrict__ row, const int* __restrict__ col,
                      const float* __restrict__ w, const float* __restrict__ dinv,
                      float* __restrict__ nrm, int E) {
    int e = blockIdx.x * blockDim.x + threadIdx.x;
    if (e < E) nrm[e] = dinv[row[e]] * w[e] * dinv[col[e]];
}

// ---------------- scatter: dst[col] += norm * src[row]; 1 wave / edge ----------------
__global__ __launch_bounds__(256)
void scatter_kernel(const int* __restrict__ row, const int* __restrict__ col,
                    const float* __restrict__ nrm, const float* __restrict__ src,
                    float* __restrict__ dst, int E) {
    int wave = (int)((blockIdx.x * blockDim.x + threadIdx.x) >> 5);
    int lane = threadIdx.x & 31;
    if (wave >= E) return;
    int r = row[wave];
    int c = col[wave];
    float nv = nrm[wave];
    const float4* s4 = (const float4*)(src + (size_t)r * C_DIM);
    float* d = dst + (size_t)c * C_DIM;
#pragma unroll
    for (int i = 0; i < 2; ++i) {
        float4 v = s4[lane + 32 * i];      // global_load_b128 of source row
        int base = (lane + 32 * i) * 4;
        atomicAdd(&d[base + 0], nv * v.x);
        atomicAdd(&d[base + 1], nv * v.y);
        atomicAdd(&d[base + 2], nv * v.z);
        atomicAdd(&d[base + 3], nv * v.w);
    }
}

// ---------------- GEMM: out[M,256] = A[M,256] * W[256,256]^T + b, fp32 WMMA ----------------
// One wave32 per 16x16 output tile, K-loop of 64 x (K=4) v_wmma_f32_16x16x4_f32.
__global__ __launch_bounds__(256)
void gemm_wmma_kernel(const float* __restrict__ A, const float* __restrict__ W,
                      const float* __restrict__ bias, float* __restrict__ out,
                      int M) {
    const int NT = C_DIM / 16;                               // 16 N-tiles
    int wave = (int)((blockIdx.x * blockDim.x + threadIdx.x) >> 5);
    int lane = threadIdx.x & 31;
    int tm = wave / NT;
    int tn = wave % NT;
    if (tm * 16 >= M) return;

    int half = lane >> 4;      // 0: lanes 0-15 (K=0,1), 1: lanes 16-31 (K=2,3)
    int l    = lane & 15;      // A: M=l ; B: N=l

    int arow = tm * 16 + l;
    if (arow >= M) arow = M - 1;                 // clamp (stores are guarded)
    const float* Ap = A + (size_t)arow * C_DIM;  // A row-major [M,256]
    const float* Wp = W + (size_t)(tn * 16 + l) * C_DIM; // B[k][n] = W[n][k]

    v8f acc = {};
    for (int k = 0; k < C_DIM; k += 4) {
        int k0 = k + half * 2;
        v2f a, b;
        a.x = Ap[k0];
        a.y = Ap[k0 + 1];
        b.x = Wp[k0];
        b.y = Wp[k0 + 1];
        acc = __builtin_amdgcn_wmma_f32_16x16x4_f32(
            /*neg_a=*/false, a, /*neg_b=*/false, b,
            /*c_mod=*/(short)0, acc, /*reuse_a=*/false, /*reuse_b=*/false);
    }

    int n  = tn * 16 + l;
    float bv = bias[n];
#pragma unroll
    for (int r = 0; r < 8; ++r) {
        int m = tm * 16 + r + half * 8;          // C/D layout: VGPR r, half-split M
        if (m < M) out[(size_t)m * C_DIM + n] = acc[r] + bv;
    }
}

// ---------------- host-side launcher ----------------
extern "C" void kernel_launch(void* const* d_in, const int* in_sizes, int n_in,
                              void* d_out, int out_size, void* d_ws, size_t ws_size,
                              hipStream_t stream) {
    const float* x     = (const float*)d_in[0];
    const int*   eidx  = (const int*)d_in[1];      // [2, E]: row then col
    const float* ew    = (const float*)d_in[2];
    const float* lin_w = (const float*)d_in[3];    // [256, 256]
    const float* lin_b = (const float*)d_in[4];    // [256]

    const int N = in_sizes[0] / C_DIM;             // 20000
    const int E = in_sizes[2];                     // 640000
    const int* row = eidx;
    const int* col = eidx + E;

    // workspace layout (floats)
    float* xA  = (float*)d_ws;
    float* xB  = xA + (size_t)N * C_DIM;
    float* deg = xB + (size_t)N * C_DIM;
    float* nrm = deg + N;

    const int B = 256;
    const size_t NC = (size_t)N * C_DIM;

    // 1) L2 row-normalize: one wave per row
    normalize_rows<<<(N * 32 + B - 1) / B, B, 0, stream>>>(x, xA, N);

    // 2) degree over targets
    zero_kernel<<<(N + B - 1) / B, B, 0, stream>>>(deg, N);
    deg_kernel<<<(E + B - 1) / B, B, 0, stream>>>(col, ew, deg, E);
    dinv_kernel<<<(N + B - 1) / B, B, 0, stream>>>(deg, N);

    // 3) per-edge symmetric norm
    edge_norm_kernel<<<(E + B - 1) / B, B, 0, stream>>>(row, col, ew, deg, nrm, E);

    // 4) hop 1: xB = P xA
    zero_kernel<<<((int)NC + B - 1) / B, B, 0, stream>>>(xB, (int)NC);
    scatter_kernel<<<(E * 32 + B - 1) / B, B, 0, stream>>>(row, col, nrm, xA, xB, E);

    // 5) hop 2: xA = P xB
    zero_kernel<<<((int)NC + B - 1) / B, B, 0, stream>>>(xA, (int)NC);
    scatter_kernel<<<(E * 32 + B - 1) / B, B, 0, stream>>>(row, col, nrm, xB, xA, E);

    // 6) linear layer via fp32 WMMA: out = xA @ W^T + b
    int mtiles = (N + 15) / 16;                    // 1250
    int waves  = mtiles * (C_DIM / 16);            // 20000 tiles
    gemm_wmma_kernel<<<(waves * 32 + B - 1) / B, B, 0, stream>>>(xA, lin_w, lin_b,
                                                                 (float*)d_out, N);
}